// ReuploadingQuantumLayer_41377714929706
// MI455X (gfx1250) — compile-verified
//
#include <hip/hip_runtime.h>

typedef float v2f __attribute__((ext_vector_type(2)));
typedef float v8f __attribute__((ext_vector_type(8)));

#define NQ  10
#define NL  4
#define DIM 1024      // 2^10 amplitudes
#define WPW 4         // waves (batch elements) per workgroup

// (r,c) <-> flat amplitude index for each 4-qubit group
// grp0: qubits 0-3  = bits 9..6 ; grp1: qubits 4-7 = bits 5..2 ; grp2: (I,I,q8,q9) = bits 3..0
__device__ __forceinline__ int map_idx(int grp, int r, int c) {
  if (grp == 0) return (r << 6) | c;
  if (grp == 1) return ((c >> 2) << 6) | (r << 2) | (c & 3);
  return (c << 4) | r;
}

// Composition of CNOT(q -> q+1 mod 10) for q = 0..9 as a GF(2)-linear index map.
// New bit(k>=1) = XOR(c0..ck) (prefix xor from MSB), new bit0(qubit0) = XOR(c1..c9).
// Linear over GF(2): perm(a ^ b) == perm(a) ^ perm(b).
__device__ __forceinline__ constexpr int cnot_perm(int i) {
  int p = i;
  p ^= p >> 1; p ^= p >> 2; p ^= p >> 4; p ^= p >> 8;   // bit j = xor of bits j..9
  return (p & 0x1FF) | (((p ^ (i >> 9)) & 1) << 9);     // fix qubit-0 bit: parity ^ c0
}

__global__ __launch_bounds__(WPW * 32)
void qsim_wmma_kernel(const float* __restrict__ x,
                      const float* __restrict__ w,
                      float* __restrict__ out, int batch) {
  __shared__ float lsRe[WPW][DIM];
  __shared__ float lsIm[WPW][DIM];
  __shared__ float lsG[WPW][NQ * 8];   // per-qubit 2x2 complex gate: 8 floats each

  const int lane = threadIdx.x & 31;
  const int wv   = threadIdx.x >> 5;
  const int b    = blockIdx.x * WPW + wv;
  if (b >= batch) return;

  float* mre = lsRe[wv];
  float* mim = lsIm[wv];
  float* sg  = lsG[wv];

  // |0...0>
  #pragma unroll
  for (int jj = 0; jj < DIM / 32; ++jj) {
    int i = jj * 32 + lane;
    mre[i] = (i == 0) ? 1.0f : 0.0f;
    mim[i] = 0.0f;
  }

  const int m     = lane & 15;          // WMMA A-fragment row / C-fragment column
  const int khalf = (lane >> 4) << 1;   // 0 or 2 : A-fragment K base within a K=4 chunk
  const int mb8   = (lane >> 4) << 3;   // C/D fragment row base (0 or 8)
  const int pmb8  = cnot_perm(mb8);     // runtime (lane-dep) part of permuted store addr

  #pragma unroll 1
  for (int l = 0; l < NL; ++l) {
    // ---- per-qubit gates U = RY(w1) RZ(w0) RY(atan(x)*pi), lanes 0..9 ----
    if (lane < NQ) {
      float xv = x[b * (NL * NQ) + l * NQ + lane];
      float w0 = w[(l * NQ + lane) * 2 + 0];
      float w1 = w[(l * NQ + lane) * 2 + 1];
      float ah = atanf(xv) * 1.5707963267948966f;   // a/2 = atan(x)*pi/2
      float ca = cosf(ah), sa = sinf(ah);
      float ph = 0.5f * w0;
      float cp = cosf(ph), sp = sinf(ph);
      float wh = 0.5f * w1;
      float cw = cosf(wh), sw = sinf(wh);
      float A = cw * ca, B = sw * sa, C = cw * sa, D = sw * ca;
      float* g = sg + lane * 8;
      g[0] = (A - B) * cp;  g[1] = -(A + B) * sp;   // u00
      g[2] = -(C + D) * cp; g[3] = (C - D) * sp;    // u01
      g[4] = (C + D) * cp;  g[5] = (C - D) * sp;    // u10
      g[6] = (A - B) * cp;  g[7] = (A + B) * sp;    // u11
    }

    // ---- three fused 16x16 complex gate applications via WMMA ----
    #pragma unroll
    for (int grp = 0; grp < 3; ++grp) {
      // constituent 2x2 gates into registers (identity padding for grp 2, pos 0/1)
      float u00r[4], u00i[4], u01r[4], u01i[4], u10r[4], u10i[4], u11r[4], u11i[4];
      #pragma unroll
      for (int p = 0; p < 4; ++p) {
        if (grp == 2 && p < 2) {
          u00r[p] = 1.f; u00i[p] = 0.f; u01r[p] = 0.f; u01i[p] = 0.f;
          u10r[p] = 0.f; u10i[p] = 0.f; u11r[p] = 1.f; u11i[p] = 0.f;
        } else {
          int q = (grp == 2) ? (6 + p) : (grp * 4 + p);
          const float* g = sg + q * 8;
          u00r[p] = g[0]; u00i[p] = g[1]; u01r[p] = g[2]; u01i[p] = g[3];
          u10r[p] = g[4]; u10i[p] = g[5]; u11r[p] = g[6]; u11i[p] = g[7];
        }
      }
      // A-matrix fragments of G = kron of 4 gates (ISA 16x4 f32 layout)
      v2f Gr[4], Gi[4], Gin[4];
      #pragma unroll
      for (int j = 0; j < 4; ++j) {
        #pragma unroll
        for (int t = 0; t < 2; ++t) {
          int k = 4 * j + khalf + t;
          float re = 1.f, im = 0.f;
          #pragma unroll
          for (int p = 0; p < 4; ++p) {
            int mbit = (m >> (3 - p)) & 1;
            int kbit = (k >> (3 - p)) & 1;
            float ur = mbit ? (kbit ? u11r[p] : u10r[p]) : (kbit ? u01r[p] : u00r[p]);
            float ui = mbit ? (kbit ? u11i[p] : u10i[p]) : (kbit ? u01i[p] : u00i[p]);
            float nr = re * ur - im * ui;
            im = re * ui + im * ur;
            re = nr;
          }
          Gr[j][t] = re; Gi[j][t] = im; Gin[j][t] = -im;
        }
      }

      if (grp < 2) {
        // ---- tile-local in-place update (reads and writes hit the same (r,c) block) ----
        #pragma unroll
        for (int tc = 0; tc < 4; ++tc) {
          int col = tc * 16 + m;
          v2f Bre[4], Bim[4];
          #pragma unroll
          for (int j = 0; j < 4; ++j) {
            int k0 = 4 * j + khalf;
            int i0 = map_idx(grp, k0,     col);
            int i1 = map_idx(grp, k0 + 1, col);
            Bre[j][0] = mre[i0]; Bre[j][1] = mre[i1];
            Bim[j][0] = mim[i0]; Bim[j][1] = mim[i1];
          }
          v8f aR, aI;
          #pragma unroll
          for (int e = 0; e < 8; ++e) { aR[e] = 0.f; aI[e] = 0.f; }
          #pragma unroll
          for (int j = 0; j < 4; ++j)
            aR = __builtin_amdgcn_wmma_f32_16x16x4_f32(false, Gr[j],  false, Bre[j], (short)0, aR, false, false);
          #pragma unroll
          for (int j = 0; j < 4; ++j)
            aR = __builtin_amdgcn_wmma_f32_16x16x4_f32(false, Gin[j], false, Bim[j], (short)0, aR, false, false);
          #pragma unroll
          for (int j = 0; j < 4; ++j)
            aI = __builtin_amdgcn_wmma_f32_16x16x4_f32(false, Gr[j],  false, Bim[j], (short)0, aI, false, false);
          #pragma unroll
          for (int j = 0; j < 4; ++j)
            aI = __builtin_amdgcn_wmma_f32_16x16x4_f32(false, Gi[j],  false, Bre[j], (short)0, aI, false, false);
          #pragma unroll
          for (int v = 0; v < 8; ++v) {
            int i = map_idx(grp, mb8 + v, col);
            mre[i] = aR[v]; mim[i] = aI[v];
          }
        }
      } else {
        // ---- group 2 with the CNOT-ring permutation folded into the writeback ----
        // Permuted stores scatter globally, so preload ALL fragments before any store.
        v2f Bre[16], Bim[16];   // [tc*4 + j]
        #pragma unroll
        for (int tc = 0; tc < 4; ++tc) {
          int col = tc * 16 + m;
          #pragma unroll
          for (int j = 0; j < 4; ++j) {
            int k0 = 4 * j + khalf;
            int i0 = map_idx(2, k0,     col);
            int i1 = map_idx(2, k0 + 1, col);
            Bre[tc * 4 + j][0] = mre[i0]; Bre[tc * 4 + j][1] = mre[i1];
            Bim[tc * 4 + j][0] = mim[i0]; Bim[tc * 4 + j][1] = mim[i1];
          }
        }
        #pragma unroll
        for (int tc = 0; tc < 4; ++tc) {
          v8f aR, aI;
          #pragma unroll
          for (int e = 0; e < 8; ++e) { aR[e] = 0.f; aI[e] = 0.f; }
          #pragma unroll
          for (int j = 0; j < 4; ++j)
            aR = __builtin_amdgcn_wmma_f32_16x16x4_f32(false, Gr[j],  false, Bre[tc * 4 + j], (short)0, aR, false, false);
          #pragma unroll
          for (int j = 0; j < 4; ++j)
            aR = __builtin_amdgcn_wmma_f32_16x16x4_f32(false, Gin[j], false, Bim[tc * 4 + j], (short)0, aR, false, false);
          #pragma unroll
          for (int j = 0; j < 4; ++j)
            aI = __builtin_amdgcn_wmma_f32_16x16x4_f32(false, Gr[j],  false, Bim[tc * 4 + j], (short)0, aI, false, false);
          #pragma unroll
          for (int j = 0; j < 4; ++j)
            aI = __builtin_amdgcn_wmma_f32_16x16x4_f32(false, Gi[j],  false, Bre[tc * 4 + j], (short)0, aI, false, false);
          // store addr = perm((col<<4) | (mb8^v)) = perm(col<<4) ^ perm(mb8) ^ perm(v)
          int pbase = cnot_perm((tc * 16 + m) << 4) ^ pmb8;
          #pragma unroll
          for (int v = 0; v < 8; ++v) {
            int d = pbase ^ cnot_perm(v);   // cnot_perm(v) folds to a constant
            mre[d] = aR[v]; mim[d] = aI[v];
          }
        }
      }
    }
  }

  // ---- <Z_q> = sum_i (+/-)|psi_i|^2, sign from bit (9-q) ----
  float acc[NQ];
  #pragma unroll
  for (int q = 0; q < NQ; ++q) acc[q] = 0.f;
  #pragma unroll 1
  for (int jj = 0; jj < 32; ++jj) {
    int i = jj * 32 + lane;
    float re = mre[i], im = mim[i];
    float pr = re * re + im * im;
    #pragma unroll
    for (int q = 0; q < NQ; ++q)
      acc[q] += ((i >> (9 - q)) & 1) ? -pr : pr;
  }
  #pragma unroll
  for (int q = 0; q < NQ; ++q) {
    float v = acc[q];
    for (int off = 16; off > 0; off >>= 1)
      v += __shfl_xor(v, off, 32);
    acc[q] = v;
  }
  if (lane == 0) {
    #pragma unroll
    for (int q = 0; q < NQ; ++q) out[b * NQ + q] = acc[q];
  }
}

extern "C" void kernel_launch(void* const* d_in, const int* in_sizes, int n_in,
                              void* d_out, int out_size, void* d_ws, size_t ws_size,
                              hipStream_t stream) {
  const float* x = (const float*)d_in[0];   // [B, 40] f32
  const float* w = (const float*)d_in[1];   // [4, 10, 2] f32
  float* out = (float*)d_out;               // [B, 10] f32
  int batch = in_sizes[0] / (NL * NQ);
  int blocks = (batch + WPW - 1) / WPW;
  qsim_wmma_kernel<<<blocks, WPW * 32, 0, stream>>>(x, w, out, batch);
}